// GatNet_64991445123421
// MI455X (gfx1250) — compile-verified
//
#include <hip/hip_runtime.h>
#include <hip/hip_bf16.h>
#include <math.h>

#define N_NODES 100000
#define N_EDGES 1600000
#define N_GRAPHS 1024

typedef float v2f __attribute__((ext_vector_type(2)));
typedef float v8f __attribute__((ext_vector_type(8)));

// ---------------------------------------------------------------------------
// Tiled WMMA GEMM: C[N x 128] = A[N x K] @ W[K x 128], K in {64,128}.
// Requires N % 16 == 0 (true: 100000). One 256-thread block = 8 waves;
// W staged in LDS (<= 64KB); each wave owns a 16-row strip and keeps all
// 8 column-tile f32 accumulators in VGPRs. Per K-step: 1 global b64 load
// (A frag, reused across 8 tiles) + 16 ds loads + 8 v_wmma_f32_16x16x4_f32.
// ---------------------------------------------------------------------------
template <int K>
__global__ __launch_bounds__(256) void gemm_wmma_tiled(
    const float* __restrict__ A, const float* __restrict__ Wg,
    float* __restrict__ C, int N) {
    constexpr int M = 128;
    __shared__ float Wlds[K * M];   // 32KB (K=64) or 64KB (K=128)

    // cooperative stage of the weight matrix
    for (int i = threadIdx.x; i < K * M; i += 256) Wlds[i] = Wg[i];
    __syncthreads();

    const int wave = threadIdx.x >> 5;
    const int lane = threadIdx.x & 31;
    const int lo16 = lane & 15;
    const int hi   = lane >> 4;                 // K-halves split across half-waves
    const int row0 = (blockIdx.x * 8 + wave) * 16;
    if (row0 >= N) return;                      // wave-uniform; after the only barrier

    v8f acc[8];
    const v8f zero = {};
#pragma unroll
    for (int t = 0; t < 8; ++t) acc[t] = zero;

    const float* arow = A + (long)(row0 + lo16) * K;

    for (int k0 = 0; k0 < K; k0 += 4) {
        const int ka = k0 + hi * 2;
        // A 16x4 fragment: lane m=lo16, VGPR pair = K {ka, ka+1}; contiguous 8B load
        const v2f a = *(const v2f*)(arow + ka);
#pragma unroll
        for (int t = 0; t < 8; ++t) {
            v2f b;
            b.x = Wlds[ka * M + t * 16 + lo16];
            b.y = Wlds[(ka + 1) * M + t * 16 + lo16];
            acc[t] = __builtin_amdgcn_wmma_f32_16x16x4_f32(
                false, a, false, b, (short)0, acc[t], false, false);
        }
    }

    // C/D layout: VGPR j -> row (row0 + j + hi*8), col = t*16 + lo16
#pragma unroll
    for (int t = 0; t < 8; ++t) {
#pragma unroll
        for (int j = 0; j < 8; ++j) {
            C[(long)(row0 + j + hi * 8) * M + t * 16 + lo16] = acc[t][j];
        }
    }
}

// Layer-1 linear: [N,2] @ [2,64] -- 2 FMAs per output, pure streaming.
__global__ void gemm2_kernel(const float* __restrict__ x,
                             const float* __restrict__ W,
                             float* __restrict__ out, int N) {
    long t = (long)blockIdx.x * blockDim.x + threadIdx.x;
    if (t >= (long)N * 64) return;
    const int  c = (int)(t & 63);
    const long n = t >> 6;
    out[t] = x[n * 2] * W[c] + x[n * 2 + 1] * W[64 + c];
}

// ---------------------------------------------------------------------------
// Utility fill
// ---------------------------------------------------------------------------
__global__ void fill_kernel(float* __restrict__ p, float v, long n) {
    long t = (long)blockIdx.x * blockDim.x + threadIdx.x;
    if (t < n) p[t] = v;
}

// ---------------------------------------------------------------------------
// Attention coefficients: e_src[n,h] = <h[n,h,:], a_src[h,:]>, likewise e_dst
// ---------------------------------------------------------------------------
__global__ void attn_score_kernel(const float* __restrict__ h,
                                  const float* __restrict__ a_src,
                                  const float* __restrict__ a_dst,
                                  float* __restrict__ es,
                                  float* __restrict__ ed,
                                  int N, int H, int F) {
    long t = (long)blockIdx.x * blockDim.x + threadIdx.x;
    if (t >= (long)N * H) return;
    const int  hh = (int)(t % H);
    const long n  = t / H;
    const float* hp = h + n * (long)(H * F) + hh * F;
    float s1 = 0.f, s2 = 0.f;
    for (int f = 0; f < F; ++f) {
        const float v = hp[f];
        s1 += v * a_src[hh * F + f];
        s2 += v * a_dst[hh * F + f];
    }
    es[t] = s1;
    ed[t] = s2;
}

// Float atomic max via monotone signed-max / unsigned-min on bit patterns.
__device__ __forceinline__ void atomicMaxFloat(float* addr, float val) {
    if (val >= 0.0f) {
        atomicMax((int*)addr, __float_as_int(val));
    } else {
        atomicMin((unsigned int*)addr, (unsigned int)__float_as_int(val));
    }
}

__device__ __forceinline__ float lrelu(float v) {
    return v > 0.0f ? v : 0.2f * v;
}

// ---------------------------------------------------------------------------
// Edge pass 1: segment max over dst of leaky_relu(e_src[src]+e_dst[dst])
// Edges [0,E) from edge_index; [E, E+N) are implicit self-loops.
// ---------------------------------------------------------------------------
__global__ void edge_max_kernel(const int* __restrict__ ei,
                                const float* __restrict__ es,
                                const float* __restrict__ ed,
                                float* __restrict__ mx,
                                int E, int N, int H) {
    long t = (long)blockIdx.x * blockDim.x + threadIdx.x;
    const long total = (long)(E + N) * H;
    if (t >= total) return;
    const int  hh = (int)(t % H);
    const long e  = t / H;
    int s, d;
    if (e < E) { s = ei[e]; d = ei[(long)E + e]; } else { s = d = (int)(e - E); }
    const float v = lrelu(es[(long)s * H + hh] + ed[(long)d * H + hh]);
    atomicMaxFloat(&mx[(long)d * H + hh], v);
}

// Edge pass 2: den[dst,h] += exp(e - mx[dst,h])
__global__ void edge_den_kernel(const int* __restrict__ ei,
                                const float* __restrict__ es,
                                const float* __restrict__ ed,
                                const float* __restrict__ mx,
                                float* __restrict__ den,
                                int E, int N, int H) {
    long t = (long)blockIdx.x * blockDim.x + threadIdx.x;
    const long total = (long)(E + N) * H;
    if (t >= total) return;
    const int  hh = (int)(t % H);
    const long e  = t / H;
    int s, d;
    if (e < E) { s = ei[e]; d = ei[(long)E + e]; } else { s = d = (int)(e - E); }
    const float v = lrelu(es[(long)s * H + hh] + ed[(long)d * H + hh]);
    atomicAdd(&den[(long)d * H + hh], __expf(v - mx[(long)d * H + hh]));
}

// Edge pass 3: out[dst, :] += alpha * h[src, :]; one float4 of h per thread
__global__ void edge_agg_kernel(const int* __restrict__ ei,
                                const float* __restrict__ hsrc,
                                const float* __restrict__ es,
                                const float* __restrict__ ed,
                                const float* __restrict__ mx,
                                const float* __restrict__ den,
                                float* __restrict__ out,
                                int E, int N, int H, int F) {
    const int HF = H * F;
    const int C  = HF >> 2;  // 4-feature chunks
    long t = (long)blockIdx.x * blockDim.x + threadIdx.x;
    const long total = (long)(E + N) * C;
    if (t >= total) return;
    const int  c = (int)(t % C);
    const long e = t / C;
    int s, d;
    if (e < E) { s = ei[e]; d = ei[(long)E + e]; } else { s = d = (int)(e - E); }
    const int hh = (c * 4) / F;       // all 4 features share one head (F is 8 or 16)
    const float v     = lrelu(es[(long)s * H + hh] + ed[(long)d * H + hh]);
    const float alpha = __expf(v - mx[(long)d * H + hh]) / den[(long)d * H + hh];
    const float4 hv = *(const float4*)(hsrc + (long)s * HF + c * 4);  // b128 load
    float* op = out + (long)d * HF + c * 4;
    atomicAdd(&op[0], alpha * hv.x);
    atomicAdd(&op[1], alpha * hv.y);
    atomicAdd(&op[2], alpha * hv.z);
    atomicAdd(&op[3], alpha * hv.w);
}

// out = elu(out + bias)
__global__ void bias_elu_kernel(float* __restrict__ h,
                                const float* __restrict__ b,
                                long n, int HF) {
    long t = (long)blockIdx.x * blockDim.x + threadIdx.x;
    if (t >= n * HF) return;
    const int c = (int)(t % HF);
    const float v = h[t] + b[c];
    h[t] = v > 0.0f ? v : expm1f(v);
}

// Global mean-pool accumulation
__global__ void pool_kernel(const float* __restrict__ h,
                            const int* __restrict__ batch,
                            float* __restrict__ psum,
                            float* __restrict__ pcnt,
                            int N) {
    long t = (long)blockIdx.x * blockDim.x + threadIdx.x;
    if (t >= (long)N * 128) return;
    const int c = (int)(t % 128);
    const int n = (int)(t / 128);
    const int g = batch[n];
    atomicAdd(&psum[(long)g * 128 + c], h[t]);
    if (c == 0) atomicAdd(&pcnt[g], 1.0f);
}

// BN (eval) + fc1 + relu + fc2 + log_softmax; one 128-thread block per graph
__global__ void head_kernel(const float* __restrict__ psum,
                            const float* __restrict__ pcnt,
                            const float* __restrict__ bn_g,
                            const float* __restrict__ bn_b,
                            const float* __restrict__ bn_m,
                            const float* __restrict__ bn_v,
                            const float* __restrict__ w1,
                            const float* __restrict__ b1,
                            const float* __restrict__ w2,
                            const float* __restrict__ b2,
                            float* __restrict__ out) {
    const int g = blockIdx.x;
    const int t = threadIdx.x;   // 128 threads
    __shared__ float gl[128];
    __shared__ float y1[32];
    __shared__ float y2[10];
    __shared__ float red[2];

    const float cnt = fmaxf(pcnt[g], 1.0f);
    float v = psum[(long)g * 128 + t] / cnt;
    v = (v - bn_m[t]) * rsqrtf(bn_v[t] + 1e-5f) * bn_g[t] + bn_b[t];
    gl[t] = v;
    __syncthreads();

    if (t < 32) {
        float acc = b1[t];
        for (int i = 0; i < 128; ++i) acc += gl[i] * w1[i * 32 + t];
        y1[t] = fmaxf(acc, 0.0f);
    }
    __syncthreads();

    if (t < 10) {
        float acc = b2[t];
        for (int j = 0; j < 32; ++j) acc += y1[j] * w2[j * 10 + t];
        y2[t] = acc;
    }
    __syncthreads();

    if (t == 0) {
        float mxv = y2[0];
        for (int k = 1; k < 10; ++k) mxv = fmaxf(mxv, y2[k]);
        float se = 0.f;
        for (int k = 0; k < 10; ++k) se += __expf(y2[k] - mxv);
        red[0] = mxv;
        red[1] = __logf(se);
    }
    __syncthreads();
    if (t < 10) out[(long)g * 10 + t] = y2[t] - red[0] - red[1];
}

// ---------------------------------------------------------------------------
// Host-side orchestration
// ---------------------------------------------------------------------------
static inline unsigned nblk(long n, int bs) { return (unsigned)((n + bs - 1) / bs); }

extern "C" void kernel_launch(void* const* d_in, const int* in_sizes, int n_in,
                              void* d_out, int out_size, void* d_ws, size_t ws_size,
                              hipStream_t stream) {
    const float* x     = (const float*)d_in[0];
    const int*   ei    = (const int*)  d_in[1];
    const int*   batch = (const int*)  d_in[2];
    const float* W[3]  = {(const float*)d_in[3], (const float*)d_in[7],  (const float*)d_in[11]};
    const float* AS[3] = {(const float*)d_in[4], (const float*)d_in[8],  (const float*)d_in[12]};
    const float* AD[3] = {(const float*)d_in[5], (const float*)d_in[9],  (const float*)d_in[13]};
    const float* BI[3] = {(const float*)d_in[6], (const float*)d_in[10], (const float*)d_in[14]};
    const float* bn_g  = (const float*)d_in[15];
    const float* bn_b  = (const float*)d_in[16];
    const float* bn_m  = (const float*)d_in[17];
    const float* bn_v  = (const float*)d_in[18];
    const float* fc1_w = (const float*)d_in[19];
    const float* fc1_b = (const float*)d_in[20];
    const float* fc2_w = (const float*)d_in[21];
    const float* fc2_b = (const float*)d_in[22];

    const long N = N_NODES;
    const int  E = N_EDGES;
    const int  H = 8;

    // workspace carve-up (floats)
    float* ws   = (float*)d_ws;
    float* bufA = ws;                      // N*128
    float* bufB = bufA + N * 128;          // N*128
    float* bufC = bufB + N * 128;          // N*128
    float* es   = bufC + N * 128;          // N*8
    float* ed   = es   + N * 8;            // N*8
    float* mx   = ed   + N * 8;            // N*8
    float* den  = mx   + N * 8;            // N*8
    float* psum = den  + N * 8;            // 1024*128
    float* pcnt = psum + (long)N_GRAPHS * 128;  // 1024

    // attention + aggregation phase shared by all layers
    auto run_attention = [&](const float* gbuf, int F, const float* asl,
                             const float* adl, const float* bl, float* obuf) {
        const int HF = H * F;
        attn_score_kernel<<<nblk(N * H, 256), 256, 0, stream>>>(
            gbuf, asl, adl, es, ed, (int)N, H, F);
        fill_kernel<<<nblk(N * H, 256), 256, 0, stream>>>(mx, -INFINITY, N * H);
        fill_kernel<<<nblk(N * H, 256), 256, 0, stream>>>(den, 0.0f, N * H);
        fill_kernel<<<nblk(N * HF, 256), 256, 0, stream>>>(obuf, 0.0f, N * HF);
        const long etH = (long)(E + N) * H;
        edge_max_kernel<<<nblk(etH, 256), 256, 0, stream>>>(ei, es, ed, mx, E, (int)N, H);
        edge_den_kernel<<<nblk(etH, 256), 256, 0, stream>>>(ei, es, ed, mx, den, E, (int)N, H);
        const long etC = (long)(E + N) * (HF / 4);
        edge_agg_kernel<<<nblk(etC, 256), 256, 0, stream>>>(
            ei, gbuf, es, ed, mx, den, obuf, E, (int)N, H, F);
        bias_elu_kernel<<<nblk(N * HF, 256), 256, 0, stream>>>(obuf, bl, N, HF);
    };

    // ---- layer 1: 2 -> 8x8 (scalar GEMM; K=2 too small for matrix core)
    gemm2_kernel<<<nblk(N * 64, 256), 256, 0, stream>>>(x, W[0], bufA, (int)N);
    run_attention(bufA, 8, AS[0], AD[0], BI[0], bufB);

    // ---- layer 2: 64 -> 8x16 (WMMA GEMM, W in LDS)
    gemm_wmma_tiled<64><<<nblk(N, 128), 256, 0, stream>>>(bufB, W[1], bufA, (int)N);
    run_attention(bufA, 16, AS[1], AD[1], BI[1], bufC);

    // ---- layer 3: 128 -> 8x16 (WMMA GEMM, W in LDS)
    gemm_wmma_tiled<128><<<nblk(N, 128), 256, 0, stream>>>(bufC, W[2], bufB, (int)N);
    run_attention(bufB, 16, AS[2], AD[2], BI[2], bufA);

    // ---- global mean pool
    fill_kernel<<<nblk((long)N_GRAPHS * 128, 256), 256, 0, stream>>>(psum, 0.0f, (long)N_GRAPHS * 128);
    fill_kernel<<<nblk(N_GRAPHS, 256), 256, 0, stream>>>(pcnt, 0.0f, N_GRAPHS);
    pool_kernel<<<nblk(N * 128, 256), 256, 0, stream>>>(bufA, batch, psum, pcnt, (int)N);

    // ---- BN + MLP head + log_softmax
    head_kernel<<<N_GRAPHS, 128, 0, stream>>>(psum, pcnt, bn_g, bn_b, bn_m, bn_v,
                                              fc1_w, fc1_b, fc2_w, fc2_b,
                                              (float*)d_out);
}